// ScaledDotProductAttention_32495722561672
// MI455X (gfx1250) — compile-verified
//
#include <hip/hip_runtime.h>

typedef __attribute__((ext_vector_type(2))) float v2f;
typedef __attribute__((ext_vector_type(8))) float v8f;

#define SEQ 1024
#define DH 64
#define NH 16
#define NB 4
#define MAXREL 16
#define NJ 33           // 2*MAXREL+1
#define ROWSTRIDE 1028  // 1024 + 4 pad floats -> rows offset by 4 banks

// One workgroup: one (b*h, 16-query-row tile). 4 waves of 32 (wave32).
__global__ __launch_bounds__(128) void relattn_kernel(
    const float* __restrict__ Q, const float* __restrict__ K,
    const float* __restrict__ V, const float* __restrict__ rel_table,
    float* __restrict__ out, float* __restrict__ attn)
{
    extern __shared__ float smem[];
    float* sScores = smem;                    // [16][ROWSTRIDE]
    float* sQrel   = smem + 16 * ROWSTRIDE;   // [16][NJ]

    const int qt   = blockIdx.x;      // query tile (0..63)
    const int bh   = blockIdx.y;      // fused batch*head (0..63)
    const int q0   = qt * 16;
    const int tid  = threadIdx.x;
    const int lane = tid & 31;
    const int wave = tid >> 5;
    const int lo   = lane & 15;       // N (or M for A-frag) within tile
    const int hi   = lane >> 4;       // selects K+2 / M+8 half

    const float* Qb = Q + (size_t)bh * SEQ * DH;
    const float* Kb = K + (size_t)bh * SEQ * DH;
    const float* Vb = V + (size_t)bh * SEQ * DH;

    const float NEG_INF = -__builtin_inff();

    // ---- Phase 0: Qrel[m][j] = dot(Q[q0+m], rel_table[j])  (16 x 33 dots) ----
    for (int idx = tid; idx < 16 * NJ; idx += 128) {
        const int m = idx / NJ, j = idx % NJ;
        const float* qr = Qb + (size_t)(q0 + m) * DH;
        const float* rr = rel_table + j * DH;
        float s = 0.f;
        #pragma unroll
        for (int d = 0; d < DH; ++d) s += qr[d] * rr[d];
        sQrel[m * NJ + j] = s;
    }
    __syncthreads();

    // ---- Preload Q fragments for WMMA f32 16x16x4 (A: lane=M, {v0,v1}={K+2hi, K+2hi+1}) ----
    v2f aq[16];
    {
        const float* qrow = Qb + (size_t)(q0 + lo) * DH + 2 * hi;
        #pragma unroll
        for (int s = 0; s < 16; ++s) { aq[s].x = qrow[4 * s]; aq[s].y = qrow[4 * s + 1]; }
    }

    const float inv_scale = 0.125f;   // 1/sqrt(64)

    // ---- Phase 1: scores = QK^T/8 + rel, causal mask, into LDS ----
    for (int kt = wave; kt < SEQ / 16; kt += 4) {
        const int k0 = kt * 16;
        if (k0 > q0 + 15) {           // wave-uniform: tile fully masked
            #pragma unroll
            for (int i = 0; i < 8; ++i)
                sScores[(8 * hi + i) * ROWSTRIDE + k0 + lo] = NEG_INF;
            continue;
        }
        v2f bk[16];
        const float* krow = Kb + (size_t)(k0 + lo) * DH + 2 * hi;
        #pragma unroll
        for (int s = 0; s < 16; ++s) { bk[s].x = krow[4 * s]; bk[s].y = krow[4 * s + 1]; }

        v8f acc = {};
        #pragma unroll
        for (int s = 0; s < 16; ++s)
            acc = __builtin_amdgcn_wmma_f32_16x16x4_f32(
                false, aq[s], false, bk[s], (short)0, acc, false, false);

        #pragma unroll
        for (int i = 0; i < 8; ++i) {
            const int m  = 8 * hi + i;       // C/D: M = i + 8*(lane>>4)
            const int gl = q0 + m;
            const int gr = k0 + lo;          // C/D: N = lane&15
            float sc;
            if (gr > gl) {
                sc = NEG_INF;
            } else {
                int rel = gr - gl;
                rel = rel < -MAXREL ? -MAXREL : (rel > MAXREL ? MAXREL : rel);
                sc = acc[i] * inv_scale + sQrel[m * NJ + rel + MAXREL];
            }
            sScores[m * ROWSTRIDE + k0 + lo] = sc;
        }
    }
    __syncthreads();

    // ---- Phase 2: row softmax; write probs to LDS and to attn output ----
    float* attn_b = attn + (size_t)bh * SEQ * SEQ;
    for (int rr = 0; rr < 4; ++rr) {
        const int row = wave + 4 * rr;
        float* srow = sScores + row * ROWSTRIDE;

        float mx = NEG_INF;
        for (int c = lane; c < SEQ; c += 32) mx = fmaxf(mx, srow[c]);
        #pragma unroll
        for (int off = 16; off >= 1; off >>= 1) mx = fmaxf(mx, __shfl_xor(mx, off, 32));

        float sum = 0.f;
        for (int c = lane; c < SEQ; c += 32) sum += __expf(srow[c] - mx);
        #pragma unroll
        for (int off = 16; off >= 1; off >>= 1) sum += __shfl_xor(sum, off, 32);

        const float inv = 1.f / sum;
        float* arow = attn_b + (size_t)(q0 + row) * SEQ;
        for (int c = lane; c < SEQ; c += 32) {
            const float p = __expf(srow[c] - mx) * inv;   // masked -> exp(-inf)=0
            srow[c] = p;
            arow[c] = p;      // coalesced across lanes
        }
    }
    __syncthreads();

    // ---- Phase 3: out = P @ V, unrolled x4 so loads clause and 4 WMMAs issue
    //      back-to-back per round (kmax = q0+16 is always a multiple of 16) ----
    {
        const int n0 = 16 * wave;
        v8f acc = {};
        const int kmax = q0 + 16;            // probs are exactly 0 beyond diagonal
        const float* prow = sScores + lo * ROWSTRIDE + 2 * hi;   // A: row=lo, col=k+2hi
        const float* vcol = Vb + (size_t)(2 * hi) * DH + n0 + lo; // B: V[k+2hi][n0+lo]

        for (int k0 = 0; k0 < kmax; k0 += 16) {
            v2f a[4], b[4];
            #pragma unroll
            for (int u = 0; u < 4; ++u) {              // batch all loads first
                const int kk = k0 + 4 * u;
                a[u].x = prow[kk];                     // ds_load_b64 (consecutive pair)
                a[u].y = prow[kk + 1];
                const float* vr = vcol + (size_t)kk * DH;
                b[u].x = vr[0];                        // V[kk+2hi  ][n0+lo]
                b[u].y = vr[DH];                       // V[kk+2hi+1][n0+lo]
            }
            #pragma unroll
            for (int u = 0; u < 4; ++u)
                acc = __builtin_amdgcn_wmma_f32_16x16x4_f32(
                    false, a[u], false, b[u], (short)0, acc, false, false);
        }
        float* ob = out + ((size_t)bh * SEQ + q0) * DH;
        #pragma unroll
        for (int i = 0; i < 8; ++i) {
            const int m = 8 * hi + i;
            ob[(size_t)m * DH + n0 + lo] = acc[i];
        }
    }
}

extern "C" void kernel_launch(void* const* d_in, const int* in_sizes, int n_in,
                              void* d_out, int out_size, void* d_ws, size_t ws_size,
                              hipStream_t stream) {
    (void)in_sizes; (void)n_in; (void)d_ws; (void)ws_size; (void)out_size;
    const float* Q   = (const float*)d_in[0];
    const float* K   = (const float*)d_in[1];
    const float* V   = (const float*)d_in[2];
    const float* rel = (const float*)d_in[3];
    // d_in[4] is the causal mask; causality is applied analytically (tril), matching the reference.

    float* out  = (float*)d_out;                                   // [B,H,S,D]
    float* attn = out + (size_t)NB * NH * SEQ * DH;                // [B,H,S,S]

    dim3 grid(SEQ / 16, NB * NH);
    dim3 block(128);
    size_t shmem = (size_t)(16 * ROWSTRIDE + 16 * NJ) * sizeof(float);  // ~68 KB of 320 KB WGP LDS
    relattn_kernel<<<grid, block, shmem, stream>>>(Q, K, V, rel, out, attn);
}